// LSTM_76381698392466
// MI455X (gfx1250) — compile-verified
//
#include <hip/hip_runtime.h>
#include <hip/hip_bf16.h>
#include <math.h>

typedef __attribute__((ext_vector_type(2))) float v2f;
typedef __attribute__((ext_vector_type(8))) float v8f;

#define BN_EPS 1e-5f
#define LEAKY_SLOPE 0.01f
#define CHUNK 16

__device__ __forceinline__ float sigmoidf_(float x) { return 1.0f / (1.0f + expf(-x)); }

// ---------------------------------------------------------------------------
// WMMA f32 GEMM: out[t][n] = act( sum_k A[t][k] * Wt[n][k] + bias1[n] (+bias2[n]) )
// A: [T,K] row-major, Wt: [N,K] row-major (computes A @ Wt^T). One wave = 16x(16*NT).
// ---------------------------------------------------------------------------
template <int NT, bool LEAKY>
__global__ __launch_bounds__(256) void wmma_gemm(const float* __restrict__ A,
                                                 const float* __restrict__ Wt,
                                                 const float* __restrict__ bias1,
                                                 const float* __restrict__ bias2,
                                                 float* __restrict__ out,
                                                 int T, int K, int N) {
  const int lane = threadIdx.x & 31;
  const int wave = blockIdx.x * (blockDim.x >> 5) + (threadIdx.x >> 5);
  const int colGroups = N / (16 * NT);
  const int rowTile = wave / colGroups;
  if (rowTile >= (T >> 4)) return;
  const int colGroup = wave - rowTile * colGroups;
  const int t0 = rowTile << 4;
  const int n0 = colGroup * (16 * NT);
  const int l16 = lane & 15;
  const bool hi = lane >= 16;
  const int kofs = hi ? 2 : 0;  // f32 16x4 A layout: lanes 0-15 K=0,1; lanes 16-31 K=2,3

  const float* Arow = A + (size_t)(t0 + l16) * K + kofs;
  const float* Brow[NT];
#pragma unroll
  for (int nt = 0; nt < NT; ++nt)
    Brow[nt] = Wt + (size_t)(n0 + nt * 16 + l16) * K + kofs;

  v8f acc[NT];
#pragma unroll
  for (int nt = 0; nt < NT; ++nt)
#pragma unroll
    for (int e = 0; e < 8; ++e) acc[nt][e] = 0.0f;

  for (int k0 = 0; k0 < K; k0 += 4) {
    v2f a;
    a[0] = Arow[k0];
    a[1] = Arow[k0 + 1];
#pragma unroll
    for (int nt = 0; nt < NT; ++nt) {
      v2f b;
      b[0] = Brow[nt][k0];
      b[1] = Brow[nt][k0 + 1];
      acc[nt] = __builtin_amdgcn_wmma_f32_16x16x4_f32(false, a, false, b, (short)0,
                                                      acc[nt], false, false);
    }
  }

  const int rbase = t0 + (hi ? 8 : 0);
#pragma unroll
  for (int nt = 0; nt < NT; ++nt) {
    const int col = n0 + nt * 16 + l16;
    float bb = bias1 ? bias1[col] : 0.0f;
    if (bias2) bb += bias2[col];
#pragma unroll
    for (int r = 0; r < 8; ++r) {
      float v = acc[nt][r] + bb;
      if (LEAKY) v = (v >= 0.0f) ? v : LEAKY_SLOPE * v;
      out[(size_t)(rbase + r) * N + col] = v;
    }
  }
}

__global__ void progress_init(int* p) { *p = 0; }

// ---------------------------------------------------------------------------
// Pipelined 2-layer LSTM: grid = 2 persistent blocks of 512 threads.
//  block 0 (producer, layer 0): Whh0 row + Wih0 row in VGPRs, h0 in LDS,
//    2 barriers/step; release-publishes progress every CHUNK steps.
//  block 1 (consumer, layer 1): Wih1 staged once into LDS (264 KB of the WGP's
//    320 KB); per CHUNK: acquire-spin on progress, coalesced-copy h0 chunk to
//    LDS, WMMA projection chunk@Wih1^T (+biases) from LDS into LDS, then 16
//    recurrence steps with Whh1 rows in VGPRs.
// Gate order (torch): i[0:128] f[128:256] g[256:384] o[384:512].
// ---------------------------------------------------------------------------
__global__ __launch_bounds__(512, 1) void lstm_pipe(
    const float* __restrict__ x, const float* __restrict__ Wih0,
    const float* __restrict__ Whh0, const float* __restrict__ bih0,
    const float* __restrict__ bhh0, const float* __restrict__ Wih1,
    const float* __restrict__ Whh1, const float* __restrict__ bih1,
    const float* __restrict__ bhh1, float* __restrict__ hs0,
    float* __restrict__ hs1, int* __restrict__ progress, int T) {
  const int j = threadIdx.x;

  if (blockIdx.x == 0) {
    // ------------------------- producer: layer 0 -------------------------
    __shared__ __align__(16) float h_sh[128];
    __shared__ float gates[512];
    __shared__ __align__(16) float xrow[8];

    float4 hw[32];
    const float4* wrow = (const float4*)(Whh0 + (size_t)j * 128);
#pragma unroll
    for (int k = 0; k < 32; ++k) hw[k] = wrow[k];
    const float4* xwr = (const float4*)(Wih0 + (size_t)j * 8);
    float4 xw0 = xwr[0], xw1 = xwr[1];
    const float bj = bih0[j] + bhh0[j];

    if (j < 128) h_sh[j] = 0.0f;
    float c = 0.0f;

    for (int t = 0; t < T; ++t) {
      if (j < 8) xrow[j] = x[(size_t)t * 8 + j];
      __syncthreads();  // h_{t-1} and xrow ready

      float4 xv0 = ((const float4*)xrow)[0];
      float4 xv1 = ((const float4*)xrow)[1];
      float z = bj;
      z += xw0.x * xv0.x + xw0.y * xv0.y + xw0.z * xv0.z + xw0.w * xv0.w;
      z += xw1.x * xv1.x + xw1.y * xv1.y + xw1.z * xv1.z + xw1.w * xv1.w;
      const float4* h4 = (const float4*)h_sh;
#pragma unroll
      for (int k = 0; k < 32; ++k) {
        float4 hv = h4[k];
        z += hw[k].x * hv.x + hw[k].y * hv.y + hw[k].z * hv.z + hw[k].w * hv.w;
      }
      gates[j] = (j < 256 || j >= 384) ? sigmoidf_(z) : tanhf(z);
      __syncthreads();

      if (j < 128) {
        float ig = gates[j], fg = gates[128 + j], gg = gates[256 + j], og = gates[384 + j];
        c = fg * c + ig * gg;
        float h = og * tanhf(c);
        h_sh[j] = h;
        hs0[(size_t)t * 128 + j] = h;
      }
      if ((t & (CHUNK - 1)) == (CHUNK - 1)) {
        __threadfence();   // release: each thread's hs0 stores reach agent scope
        __syncthreads();   // all threads fenced before publish
        if (j == 0) atomicExch(progress, t + 1);
      }
    }
  } else {
    // ------------------------- consumer: layer 1 -------------------------
    __shared__ __align__(16) float B_sh[512 * 132];     // Wih1 staged; padded stride
    __shared__ __align__(16) float h0_sh[CHUNK * 132];  // padded: conflict-free A frags
    __shared__ __align__(16) float pre_sh[CHUNK * 516]; // padded: conflict-free stores
    __shared__ __align__(16) float h1_sh[128];
    __shared__ float gates[512];

    float4 hw[32];
    const float4* wrow = (const float4*)(Whh1 + (size_t)j * 128);
#pragma unroll
    for (int k = 0; k < 32; ++k) hw[k] = wrow[k];

    const int lane = j & 31;
    const int wv = j >> 5;        // wave 0..15 -> cols [wv*32, wv*32+32)
    const int l16 = lane & 15;
    const bool hi = lane >= 16;
    const int kofs = hi ? 2 : 0;
    const int n0 = wv * 32;
    const float bias_s0 = bih1[n0 + l16] + bhh1[n0 + l16];
    const float bias_s1 = bih1[n0 + 16 + l16] + bhh1[n0 + 16 + l16];

    // Stage Wih1 row j (= B column j) into LDS once; 16B-aligned (132*4 % 16 == 0).
    {
      const float4* src = (const float4*)(Wih1 + (size_t)j * 128);
      float4* dst = (float4*)(B_sh + j * 132);
#pragma unroll
      for (int q = 0; q < 32; ++q) dst[q] = src[q];
    }
    const float* BL0 = B_sh + (n0 + l16) * 132 + kofs;
    const float* BL1 = B_sh + (n0 + 16 + l16) * 132 + kofs;

    if (j < 128) h1_sh[j] = 0.0f;
    float c = 0.0f;

    // chunk copy indexing: wave wv copies row wv, lane picks a 16B segment
    const int cr = wv;             // 0..15
    const int ck = lane * 4;       // 0..124

    for (int t0 = 0; t0 < T; t0 += CHUNK) {
      // keep all in-loop loads in the loop (no LICM hoist -> no VGPR spills)
      asm volatile("" ::: "memory");

      // 1) acquire chunk from producer
      if (j == 0) {
        while (atomicAdd(progress, 0) < t0 + CHUNK) __builtin_amdgcn_s_sleep(1);
      }
      __syncthreads();
      __threadfence();  // acquire at agent scope: invalidate WGP$ before reads

      // 2) copy h0 chunk into LDS (coalesced 16B loads, single wait)
      {
        const float4 v = *(const float4*)(hs0 + (size_t)(t0 + cr) * 128 + ck);
        h0_sh[cr * 132 + ck + 0] = v.x;
        h0_sh[cr * 132 + ck + 1] = v.y;
        h0_sh[cr * 132 + ck + 2] = v.z;
        h0_sh[cr * 132 + ck + 3] = v.w;
      }
      __syncthreads();

      // 3) WMMA projection: pre = h0_chunk[16x128] @ Wih1^T (+ biases) -> LDS
      v8f acc0, acc1;
#pragma unroll
      for (int e = 0; e < 8; ++e) { acc0[e] = 0.0f; acc1[e] = 0.0f; }
      for (int k0 = 0; k0 < 128; k0 += 4) {
        v2f a;
        a[0] = h0_sh[l16 * 132 + k0 + kofs];
        a[1] = h0_sh[l16 * 132 + k0 + kofs + 1];
        v2f b0, b1;
        b0[0] = BL0[k0]; b0[1] = BL0[k0 + 1];
        b1[0] = BL1[k0]; b1[1] = BL1[k0 + 1];
        acc0 = __builtin_amdgcn_wmma_f32_16x16x4_f32(false, a, false, b0, (short)0, acc0, false, false);
        acc1 = __builtin_amdgcn_wmma_f32_16x16x4_f32(false, a, false, b1, (short)0, acc1, false, false);
      }
      {
        const int rbase = hi ? 8 : 0;
#pragma unroll
        for (int r = 0; r < 8; ++r) {
          pre_sh[(rbase + r) * 516 + n0 + l16] = acc0[r] + bias_s0;
          pre_sh[(rbase + r) * 516 + n0 + 16 + l16] = acc1[r] + bias_s1;
        }
      }
      __syncthreads();

      // 4) 16 recurrence steps from LDS pre
      for (int s = 0; s < CHUNK; ++s) {
        float z = pre_sh[s * 516 + j];
        const float4* h4 = (const float4*)h1_sh;
#pragma unroll
        for (int k = 0; k < 32; ++k) {
          float4 hv = h4[k];
          z += hw[k].x * hv.x + hw[k].y * hv.y + hw[k].z * hv.z + hw[k].w * hv.w;
        }
        gates[j] = (j < 256 || j >= 384) ? sigmoidf_(z) : tanhf(z);
        __syncthreads();
        if (j < 128) {
          float ig = gates[j], fg = gates[128 + j], gg = gates[256 + j], og = gates[384 + j];
          c = fg * c + ig * gg;
          float h = og * tanhf(c);
          h1_sh[j] = h;
          hs1[(size_t)(t0 + s) * 128 + j] = h;
        }
        __syncthreads();  // h1 ready for next step / next phase
      }
    }
  }
}

// ---------------------------------------------------------------------------
// Deterministic two-stage mean/var. C in {16,32,64}; grid = 256 x 256 threads.
// ---------------------------------------------------------------------------
__global__ __launch_bounds__(256) void stats_partial(const float* __restrict__ y,
                                                     int rowsPerBlock, int C,
                                                     float2* __restrict__ part) {
  __shared__ float2 sd[256];
  const int tid = threadIdx.x;
  const int R = 256 / C;
  const int c = tid & (C - 1);
  const int rr = tid / C;
  const size_t r0 = (size_t)blockIdx.x * rowsPerBlock;
  float s = 0.0f, ss = 0.0f;
  for (int r = rr; r < rowsPerBlock; r += R) {
    float v = y[(r0 + r) * C + c];
    s += v;
    ss += v * v;
  }
  sd[tid] = make_float2(s, ss);
  __syncthreads();
  for (int off = 128; off >= C; off >>= 1) {
    if (tid < off) {
      sd[tid].x += sd[tid + off].x;
      sd[tid].y += sd[tid + off].y;
    }
    __syncthreads();
  }
  if (tid < C) part[blockIdx.x * C + tid] = sd[tid];
}

// Stage 2 + BN fold: W'[n][k] = W[n][k]*scale_k, b'[n] = b[n] + sum_k shift_k*W[n][k].
__global__ __launch_bounds__(256) void fold_bn(const float2* __restrict__ part, int nPart,
                                               int C, float Tf, const float* __restrict__ g,
                                               const float* __restrict__ be,
                                               const float* __restrict__ Wn,
                                               const float* __restrict__ bn,
                                               float* __restrict__ Wp, float* __restrict__ bp,
                                               int Nout) {
  __shared__ float scale[64];
  __shared__ float shift[64];
  const int tid = threadIdx.x;
  if (tid < C) {
    float s = 0.0f, ss = 0.0f;
    for (int b = 0; b < nPart; ++b) {  // fixed order -> deterministic
      float2 p = part[b * C + tid];
      s += p.x;
      ss += p.y;
    }
    float m = s / Tf;
    float v = ss / Tf - m * m;  // biased variance (matches jnp.var)
    float sc = g[tid] / sqrtf(v + BN_EPS);
    scale[tid] = sc;
    shift[tid] = be[tid] - m * sc;
  }
  __syncthreads();
  for (int e = tid; e < Nout * C; e += blockDim.x) Wp[e] = Wn[e] * scale[e % C];
  if (tid < Nout) {
    float acc = bn[tid];
    for (int k = 0; k < C; ++k) acc += shift[k] * Wn[tid * C + k];
    bp[tid] = acc;
  }
}

// Final head: out[t] = dot16(y3[t], Wop) + bop (BN3 folded into Wop/bop).
__global__ __launch_bounds__(256) void final_out(const float* __restrict__ y3,
                                                 const float* __restrict__ Wop,
                                                 const float* __restrict__ bop,
                                                 float* __restrict__ out, int T) {
  const int t = blockIdx.x * blockDim.x + threadIdx.x;
  if (t >= T) return;
  const float4* yr = (const float4*)(y3 + (size_t)t * 16);
  const float4* w = (const float4*)Wop;
  float s = bop[0];
#pragma unroll
  for (int q = 0; q < 4; ++q) {
    float4 a = yr[q], b = w[q];
    s += a.x * b.x + a.y * b.y + a.z * b.z + a.w * b.w;
  }
  out[t] = s;
}

// ---------------------------------------------------------------------------
extern "C" void kernel_launch(void* const* d_in, const int* in_sizes, int n_in,
                              void* d_out, int out_size, void* d_ws, size_t ws_size,
                              hipStream_t stream) {
  const float* x    = (const float*)d_in[0];
  const float* Wih0 = (const float*)d_in[1];
  const float* Whh0 = (const float*)d_in[2];
  const float* bih0 = (const float*)d_in[3];
  const float* bhh0 = (const float*)d_in[4];
  const float* Wih1 = (const float*)d_in[5];
  const float* Whh1 = (const float*)d_in[6];
  const float* bih1 = (const float*)d_in[7];
  const float* bhh1 = (const float*)d_in[8];
  const float* w1 = (const float*)d_in[9];
  const float* b1 = (const float*)d_in[10];
  const float* g1 = (const float*)d_in[11];
  const float* be1 = (const float*)d_in[12];
  const float* w2 = (const float*)d_in[13];
  const float* b2 = (const float*)d_in[14];
  const float* g2 = (const float*)d_in[15];
  const float* be2 = (const float*)d_in[16];
  const float* w3 = (const float*)d_in[17];
  const float* b3 = (const float*)d_in[18];
  const float* g3 = (const float*)d_in[19];
  const float* be3 = (const float*)d_in[20];
  const float* wo = (const float*)d_in[21];
  const float* bo = (const float*)d_in[22];

  const int T = in_sizes[0] / 8;  // 131072
  char* ws = (char*)d_ws;

  // Workspace (peak = 2 * T*128*4 = 128 MB for T=131072), aggressive reuse:
  float* hs0 = (float*)ws;                                      // [T,128] (dead after pipe)
  float* hs1 = (float*)(ws + (size_t)T * 128 * sizeof(float));  // [T,128]
  float* y1 = (float*)ws;                                       // [T,64]  reuses hs0 slot
  float* y2 = (float*)(ws + (size_t)T * 64 * sizeof(float));    // [T,32]
  float* y3 = (float*)(ws + (size_t)T * 96 * sizeof(float));    // [T,16]
  float2* part = (float2*)(ws + (size_t)T * 112 * sizeof(float));
  int* progress = (int*)(ws + (size_t)T * 112 * sizeof(float)); // only live during pipe
  float* W2p = (float*)((char*)part + 256 * 64 * sizeof(float2));
  float* b2p = W2p + 32 * 64;
  float* W3p = b2p + 32;
  float* b3p = W3p + 16 * 32;
  float* Wop = b3p + 16;
  float* bop = Wop + 16;

  // 1) reset pipeline progress, then run both LSTM layers concurrently (2 WGPs)
  progress_init<<<1, 1, 0, stream>>>(progress);
  lstm_pipe<<<2, 512, 0, stream>>>(x, Wih0, Whh0, bih0, bhh0, Wih1, Whh1, bih1, bhh1,
                                   hs0, hs1, progress, T);

  // 2) y1 = leaky(hs1 @ w1^T + b1)
  wmma_gemm<4, true><<<(T / 16) / 8, 256, 0, stream>>>(hs1, w1, b1, nullptr, y1, T, 128, 64);

  // 3) BN1 stats + fold into (w2,b2) -> (W2p,b2p)
  stats_partial<<<256, 256, 0, stream>>>(y1, T / 256, 64, part);
  fold_bn<<<1, 256, 0, stream>>>(part, 256, 64, (float)T, g1, be1, w2, b2, W2p, b2p, 32);

  // 4) y2 = leaky(y1 @ W2p^T + b2p)
  wmma_gemm<2, true><<<(T / 16) / 8, 256, 0, stream>>>(y1, W2p, b2p, nullptr, y2, T, 64, 32);

  // 5) BN2 stats + fold into (w3,b3) -> (W3p,b3p)
  stats_partial<<<256, 256, 0, stream>>>(y2, T / 256, 32, part);
  fold_bn<<<1, 256, 0, stream>>>(part, 256, 32, (float)T, g2, be2, w3, b3, W3p, b3p, 16);

  // 6) y3 = leaky(y2 @ W3p^T + b3p)
  wmma_gemm<1, true><<<(T / 16) / 8, 256, 0, stream>>>(y2, W3p, b3p, nullptr, y3, T, 32, 16);

  // 7) BN3 stats + fold into (wo,bo) -> (Wop,bop)
  stats_partial<<<256, 256, 0, stream>>>(y3, T / 256, 16, part);
  fold_bn<<<1, 256, 0, stream>>>(part, 256, 16, (float)T, g3, be3, wo, bo, Wop, bop, 1);

  // 8) out = y3 @ Wop^T + bop
  final_out<<<T / 256, 256, 0, stream>>>(y3, Wop, bop, (float*)d_out, T);
}